// RSHN_84550726189098
// MI455X (gfx1250) — compile-verified
//
#include <hip/hip_runtime.h>
#include <hip/hip_bf16.h>
#include <math.h>

typedef __attribute__((ext_vector_type(2))) float v2f;
typedef __attribute__((ext_vector_type(8))) float v8f;

static __device__ __forceinline__ v8f wmma4(v2f a, v2f b, v8f c) {
  // V_WMMA_F32_16X16X4_F32 : D = A(16x4) x B(4x16) + C(16x16), f32 throughout
  return __builtin_amdgcn_wmma_f32_16x16x4_f32(false, a, false, b, (short)0, c,
                                               false, false);
}

// ---------------------------------------------------------------------------
// Kernel A: 2-layer AGNN on the coarsened line graph (32 nodes, 16 feats,
// 1024 edges), fused with hW = h @ W_e  ->  [32 x 64] relation table.
// Entire problem fits in LDS; one workgroup.
// ---------------------------------------------------------------------------
__global__ void __launch_bounds__(128)
agnn_hw_kernel(const float* __restrict__ cl_h, const float* __restrict__ cl_w,
               const int* __restrict__ cl_src, const int* __restrict__ cl_dst,
               const float* __restrict__ W_e,
               const float* __restrict__ b1p, const float* __restrict__ e1p,
               const float* __restrict__ b2p, const float* __restrict__ e2p,
               float* __restrict__ hW)
{
  constexpr int NCL = 32, FCL = 16, ECL = 1024, D = 64;
  __shared__ float h[NCL][FCL];
  __shared__ float nh[NCL][FCL];
  __shared__ float agg[NCL][FCL];
  __shared__ float m[NCL], s[NCL];
  __shared__ float p[ECL];
  __shared__ float ew[ECL];
  __shared__ int   esrc[ECL], edst[ECL];

  const int tid = threadIdx.x;
  for (int i = tid; i < NCL * FCL; i += blockDim.x) h[i / FCL][i % FCL] = cl_h[i];
  for (int i = tid; i < ECL; i += blockDim.x) {
    ew[i] = cl_w[i]; esrc[i] = cl_src[i]; edst[i] = cl_dst[i];
  }
  __syncthreads();

  const float betas[2] = { b1p[0], b2p[0] };
  const float epss[2]  = { e1p[0], e2p[0] };

  for (int layer = 0; layer < 2; ++layer) {
    const float beta = betas[layer], eps = epss[layer];
    if (tid < NCL) {
      // edge_softmax(norm_by='src'): per-src max and sum of exp
      float mx = -3.402823466e38f;
      for (int i = 0; i < ECL; ++i)
        if (esrc[i] == tid) mx = fmaxf(mx, beta * ew[i]);
      float sm = 0.f;
      for (int i = 0; i < ECL; ++i)
        if (esrc[i] == tid) sm += __expf(beta * ew[i] - mx);
      m[tid] = mx; s[tid] = sm;
      // l2-normalize row tid
      float nrm = 0.f;
      for (int f = 0; f < FCL; ++f) nrm += h[tid][f] * h[tid][f];
      nrm = fmaxf(sqrtf(nrm), 1e-12f);
      for (int f = 0; f < FCL; ++f) nh[tid][f] = h[tid][f] / nrm;
      for (int f = 0; f < FCL; ++f) agg[tid][f] = 0.f;
    }
    __syncthreads();
    for (int i = tid; i < ECL; i += blockDim.x)
      p[i] = __expf(beta * ew[i] - m[esrc[i]]) / s[esrc[i]];
    __syncthreads();
    for (int i = tid; i < ECL; i += blockDim.x) {
      const int sn = esrc[i], dn = edst[i];
      const float pi = p[i];
      for (int f = 0; f < FCL; ++f) atomicAdd(&agg[dn][f], nh[sn][f] * pi);
    }
    __syncthreads();
    for (int i = tid; i < NCL * FCL; i += blockDim.x) {
      const int r = i / FCL, f = i % FCL;
      h[r][f] = fmaxf((1.f + eps) * h[r][f] + agg[r][f], 0.f);  // relu
    }
    __syncthreads();
  }

  // hW = h @ W_e   [32x16] @ [16x64]
  for (int i = tid; i < NCL * D; i += blockDim.x) {
    const int r = i / D, c = i % D;
    float acc = 0.f;
    for (int k = 0; k < FCL; ++k) acc += h[r][k] * W_e[k * D + c];
    hW[i] = acc;
  }
}

// ---------------------------------------------------------------------------
// Kernel B: edge scatter  agg[dst] += feat[src] - hW[etype]
// 16 threads per edge, 4 floats each (16B vector gathers + f32 atomics).
// ---------------------------------------------------------------------------
__global__ void __launch_bounds__(256)
edge_scatter_kernel(const float* __restrict__ feat, const float* __restrict__ hW,
                    const int* __restrict__ gsrc, const int* __restrict__ gdst,
                    const int* __restrict__ getype, float* __restrict__ agg, int E)
{
  const int gid = blockIdx.x * 256 + threadIdx.x;
  const int e = gid >> 4;
  if (e >= E) return;
  const int c = (gid & 15) << 2;
  const int s = gsrc[e], d = gdst[e], t = getype[e];
  const float4 fv = *reinterpret_cast<const float4*>(feat + (size_t)s * 64 + c);
  const float4 wv = *reinterpret_cast<const float4*>(hW + (size_t)t * 64 + c);
  float* o = agg + (size_t)d * 64 + c;
  atomicAdd(o + 0, fv.x - wv.x);
  atomicAdd(o + 1, fv.y - wv.y);
  atomicAdd(o + 2, fv.z - wv.z);
  atomicAdd(o + 3, fv.w - wv.w);
}

// ---------------------------------------------------------------------------
// Kernel C: out = tanh(feat @ W1 + agg @ W2)   [N x 64] via f32 WMMA 16x16x4.
// 256 threads = 8 waves; block tile = 32 rows x 64 cols; wave tile = 16x16.
// W1/W2 resident in LDS; feat/agg tiles staged through LDS (clamped rows so
// EXEC stays all-ones for WMMA; stores predicated).
// ---------------------------------------------------------------------------
__global__ void __launch_bounds__(256)
gemm_tanh_wmma_kernel(const float* __restrict__ feat, const float* __restrict__ agg,
                      const float* __restrict__ W1, const float* __restrict__ W2,
                      float* __restrict__ out, int N)
{
  constexpr int D = 64;
  __shared__ float sW1[D * D];
  __shared__ float sW2[D * D];
  __shared__ float sF[32 * D];
  __shared__ float sA[32 * D];
  const int tid = threadIdx.x;
  for (int i = tid; i < D * D; i += 256) { sW1[i] = W1[i]; sW2[i] = W2[i]; }

  const int R0 = blockIdx.x * 32;
  for (int i = tid; i < 32 * (D / 4); i += 256) {
    const int r = i / (D / 4), c4 = i % (D / 4);
    int gr = R0 + r; if (gr >= N) gr = N - 1;   // clamp: keep loads valid
    reinterpret_cast<float4*>(sF)[r * (D / 4) + c4] =
        reinterpret_cast<const float4*>(feat)[(size_t)gr * (D / 4) + c4];
    reinterpret_cast<float4*>(sA)[r * (D / 4) + c4] =
        reinterpret_cast<const float4*>(agg)[(size_t)gr * (D / 4) + c4];
  }
  __syncthreads();

  const int wave = tid >> 5, lane = tid & 31;
  const int half = lane >> 4, mm = lane & 15;
  const int rloc = (wave >> 2) * 16;       // 0 or 16
  const int col0 = (wave & 3) * 16;

  v8f acc = {0.f, 0.f, 0.f, 0.f, 0.f, 0.f, 0.f, 0.f};
  #pragma unroll
  for (int k0 = 0; k0 < D; k0 += 4) {
    const int ka = k0 + half * 2;          // A layout: lanes16-31 hold K+2,K+3
    v2f a1 = *reinterpret_cast<const v2f*>(&sF[(rloc + mm) * D + ka]);
    v2f b1; b1.x = sW1[ka * D + col0 + mm]; b1.y = sW1[(ka + 1) * D + col0 + mm];
    acc = wmma4(a1, b1, acc);
    v2f a2 = *reinterpret_cast<const v2f*>(&sA[(rloc + mm) * D + ka]);
    v2f b2; b2.x = sW2[ka * D + col0 + mm]; b2.y = sW2[(ka + 1) * D + col0 + mm];
    acc = wmma4(a2, b2, acc);
  }
  #pragma unroll
  for (int i = 0; i < 8; ++i) {            // C/D layout: row = i + 8*half
    const int row = R0 + rloc + half * 8 + i;
    if (row < N) out[(size_t)row * D + col0 + mm] = tanhf(acc[i]);
  }
}

// ---------------------------------------------------------------------------
// Kernel D: out = x @ W_pred + b_pred   [N x 16] via f32 WMMA 16x16x4.
// 8 waves/block -> 128 rows/block; single 16-wide col tile.
// ---------------------------------------------------------------------------
__global__ void __launch_bounds__(256)
pred_wmma_kernel(const float* __restrict__ x, const float* __restrict__ Wp,
                 const float* __restrict__ bp, float* __restrict__ out, int N)
{
  constexpr int D = 64, NC = 16;
  __shared__ float sW[D * NC];
  __shared__ float sB[NC];
  __shared__ float sX[128 * D];
  const int tid = threadIdx.x;
  for (int i = tid; i < D * NC; i += 256) sW[i] = Wp[i];
  if (tid < NC) sB[tid] = bp[tid];

  const int R0 = blockIdx.x * 128;
  for (int i = tid; i < 128 * (D / 4); i += 256) {
    const int r = i / (D / 4), c4 = i % (D / 4);
    int gr = R0 + r; if (gr >= N) gr = N - 1;
    reinterpret_cast<float4*>(sX)[r * (D / 4) + c4] =
        reinterpret_cast<const float4*>(x)[(size_t)gr * (D / 4) + c4];
  }
  __syncthreads();

  const int wave = tid >> 5, lane = tid & 31;
  const int half = lane >> 4, mm = lane & 15;
  const int rloc = wave * 16;

  v8f acc = {0.f, 0.f, 0.f, 0.f, 0.f, 0.f, 0.f, 0.f};
  #pragma unroll
  for (int k0 = 0; k0 < D; k0 += 4) {
    const int ka = k0 + half * 2;
    v2f a = *reinterpret_cast<const v2f*>(&sX[(rloc + mm) * D + ka]);
    v2f b; b.x = sW[ka * NC + mm]; b.y = sW[(ka + 1) * NC + mm];
    acc = wmma4(a, b, acc);
  }
  #pragma unroll
  for (int i = 0; i < 8; ++i) {
    const int row = R0 + rloc + half * 8 + i;
    if (row < N) out[(size_t)row * NC + mm] = acc[i] + sB[mm];
  }
}

// ---------------------------------------------------------------------------
extern "C" void kernel_launch(void* const* d_in, const int* in_sizes, int n_in,
                              void* d_out, int out_size, void* d_ws, size_t ws_size,
                              hipStream_t stream)
{
  const float* cl_h   = (const float*)d_in[0];
  const float* cl_w   = (const float*)d_in[1];
  const int*   cl_src = (const int*)d_in[2];
  const int*   cl_dst = (const int*)d_in[3];
  const int*   g_src  = (const int*)d_in[4];
  const int*   g_dst  = (const int*)d_in[5];
  const int*   g_ety  = (const int*)d_in[6];
  const float* feats  = (const float*)d_in[7];
  const float* W_e    = (const float*)d_in[8];
  const float* beta1  = (const float*)d_in[9];
  const float* eps1   = (const float*)d_in[10];
  const float* beta2  = (const float*)d_in[11];
  const float* eps2   = (const float*)d_in[12];
  const float* W1a    = (const float*)d_in[13];
  const float* W2a    = (const float*)d_in[14];
  const float* W1b    = (const float*)d_in[15];
  const float* W2b    = (const float*)d_in[16];
  const float* Wp     = (const float*)d_in[17];
  const float* bp     = (const float*)d_in[18];
  float* out = (float*)d_out;

  const int E  = in_sizes[4];
  const int NC = in_sizes[18];
  const int D  = in_sizes[17] / NC;      // 64
  const int N  = in_sizes[7] / D;        // 50000

  float* ws  = (float*)d_ws;
  float* hW  = ws;                       // 32*64
  float* agg = ws + 2048;                // N*D
  float* x1  = agg + (size_t)N * D;      // N*D
  float* x2  = x1 + (size_t)N * D;       // N*D

  agnn_hw_kernel<<<1, 128, 0, stream>>>(cl_h, cl_w, cl_src, cl_dst, W_e,
                                        beta1, eps1, beta2, eps2, hW);

  const int scatterBlocks = (int)(((long long)E * 16 + 255) / 256);
  const int gemmBlocks    = (N + 31) / 32;

  hipMemsetAsync(agg, 0, (size_t)N * D * sizeof(float), stream);
  edge_scatter_kernel<<<scatterBlocks, 256, 0, stream>>>(feats, hW, g_src, g_dst, g_ety, agg, E);
  gemm_tanh_wmma_kernel<<<gemmBlocks, 256, 0, stream>>>(feats, agg, W1a, W2a, x1, N);

  hipMemsetAsync(agg, 0, (size_t)N * D * sizeof(float), stream);
  edge_scatter_kernel<<<scatterBlocks, 256, 0, stream>>>(x1, hW, g_src, g_dst, g_ety, agg, E);
  gemm_tanh_wmma_kernel<<<gemmBlocks, 256, 0, stream>>>(x1, agg, W1b, W2b, x2, N);

  pred_wmma_kernel<<<(N + 127) / 128, 256, 0, stream>>>(x2, Wp, bp, out, N);
}